// GCNconv_90615220011128
// MI455X (gfx1250) — compile-verified
//
#include <hip/hip_runtime.h>

#define D_FEAT 128

typedef __attribute__((ext_vector_type(2))) float v2f;
typedef __attribute__((ext_vector_type(8))) float v8f;

// ---------------------------------------------------------------------------
// 1) zero the degree counters
// ---------------------------------------------------------------------------
__global__ __launch_bounds__(256) void k_zero_u32(unsigned* __restrict__ p, int n) {
  int i = blockIdx.x * blockDim.x + threadIdx.x;
  if (i < n) p[i] = 0u;
}

// ---------------------------------------------------------------------------
// 2) deg[s] += 1 for every edge source (int atomics)
// ---------------------------------------------------------------------------
__global__ __launch_bounds__(256) void k_degree(const long long* __restrict__ src,
                                                unsigned* __restrict__ deg, int E) {
  int e = blockIdx.x * blockDim.x + threadIdx.x;
  if (e < E) atomicAdd(&deg[(int)src[e]], 1u);
}

// ---------------------------------------------------------------------------
// 3) inv_norm[n] = 1 / sqrt(deg[n] + 1)
// ---------------------------------------------------------------------------
__global__ __launch_bounds__(256) void k_inv_norm(const unsigned* __restrict__ deg,
                                                  float* __restrict__ inv_norm, int n) {
  int i = blockIdx.x * blockDim.x + threadIdx.x;
  if (i < n) inv_norm[i] = 1.0f / sqrtf((float)deg[i] + 1.0f);
}

// ---------------------------------------------------------------------------
// 4) agg[n][:] = x[n][:] * inv_norm[n]   (self-loop term, float4 per thread)
// ---------------------------------------------------------------------------
__global__ __launch_bounds__(256) void k_init_agg(const float4* __restrict__ x4,
                                                  const float* __restrict__ inv_norm,
                                                  float4* __restrict__ agg4, int total4) {
  int i = blockIdx.x * blockDim.x + threadIdx.x;
  if (i >= total4) return;
  int node = i >> 5;  // 32 float4 per node row (128 floats)
  float inv = inv_norm[node];
  float4 v = x4[i];
  v.x *= inv; v.y *= inv; v.z *= inv; v.w *= inv;
  agg4[i] = v;
}

// ---------------------------------------------------------------------------
// 5) edge scatter: one wave per edge, one float4 slice per lane.
//    agg[dst][:] += x[src][:] * inv_norm[src]  via native f32 atomics (L2).
// ---------------------------------------------------------------------------
__global__ __launch_bounds__(256) void k_scatter(const float4* __restrict__ x4,
                                                 const long long* __restrict__ src,
                                                 const long long* __restrict__ dst,
                                                 const float* __restrict__ inv_norm,
                                                 float* __restrict__ agg, int E) {
  long long tid = (long long)blockIdx.x * blockDim.x + threadIdx.x;
  int e = (int)(tid >> 5);
  if (e >= E) return;
  int lane = (int)(tid & 31);
  int s = (int)src[e];
  int d = (int)dst[e];
  float inv = inv_norm[s];
  float4 v = x4[(size_t)s * (D_FEAT / 4) + lane];
  float* dp = agg + (size_t)d * D_FEAT + (lane << 2);
  unsafeAtomicAdd(dp + 0, v.x * inv);
  unsafeAtomicAdd(dp + 1, v.y * inv);
  unsafeAtomicAdd(dp + 2, v.z * inv);
  unsafeAtomicAdd(dp + 3, v.w * inv);
}

// ---------------------------------------------------------------------------
// 6) out[m][n] = sum_k (agg[m][k] * inv_norm[m]) * W[n][k] + b[n]
//    f32 WMMA 16x16x4. One wave -> one 16x16 output tile. 8 waves/block
//    cover all 8 N-tiles of the same M-tile (A rows reused from cache).
//    A layout per ISA: lane<16 holds A[m][k],A[m][k+1]; lane>=16 holds
//    A[m][k+2],A[m][k+3].  B[k][n] = W[n][k] -> contiguous float2 from W row.
// ---------------------------------------------------------------------------
__global__ __launch_bounds__(256) void k_gemm_wmma(const float* __restrict__ agg,
                                                   const float* __restrict__ inv_norm,
                                                   const float* __restrict__ W,
                                                   const float* __restrict__ bias,
                                                   float* __restrict__ out, int Nn) {
  const int lane = threadIdx.x & 31;
  const int wave = threadIdx.x >> 5;   // 0..7 -> N tile (D_OUT = 128 = 8*16)
  const int m0 = blockIdx.x << 4;
  const int n0 = wave << 4;
  const int half = lane >> 4;          // 0 or 1
  const int l16 = lane & 15;
  int arow = m0 + l16;
  if (arow >= Nn) arow = Nn - 1;       // clamp: EXEC must stay all-ones for WMMA
  const int ncol = n0 + l16;
  const int koff = half << 1;          // K sub-offset per lane half

  const float inv = inv_norm[arow];
  const v2f* __restrict__ a2 = (const v2f*)(agg + (size_t)arow * D_FEAT);
  const v2f* __restrict__ w2 = (const v2f*)(W + (size_t)ncol * D_FEAT);

  v8f c = {};
#pragma unroll
  for (int k = 0; k < D_FEAT; k += 4) {
    v2f a = a2[(k + koff) >> 1];
    a.x *= inv;
    a.y *= inv;
    v2f b = w2[(k + koff) >> 1];
    c = __builtin_amdgcn_wmma_f32_16x16x4_f32(false, a, false, b, (short)0, c,
                                              false, false);
  }

  const float bv = bias[ncol];
#pragma unroll
  for (int i = 0; i < 8; ++i) {
    int row = m0 + i + (half << 3);    // C/D layout: vgpr i -> M = i + half*8
    if (row < Nn) out[(size_t)row * D_FEAT + ncol] = c[i] + bv;
  }
}

// ---------------------------------------------------------------------------
extern "C" void kernel_launch(void* const* d_in, const int* in_sizes, int n_in,
                              void* d_out, int out_size, void* d_ws, size_t ws_size,
                              hipStream_t stream) {
  const float* x      = (const float*)d_in[0];
  const long long* ei = (const long long*)d_in[1];  // int64 edge_index [2, E]
  const float* W      = (const float*)d_in[2];
  const float* bias   = (const float*)d_in[3];
  float* out          = (float*)d_out;

  const int Nn = in_sizes[0] / D_FEAT;
  const int E  = in_sizes[1] / 2;
  const long long* src = ei;
  const long long* dst = ei + E;

  // workspace layout: deg (u32[N]) | inv_norm (f32[N]) | agg (f32[N*128])
  size_t nbytes = ((size_t)Nn * 4 + 255) & ~(size_t)255;
  char* ws = (char*)d_ws;
  unsigned* deg   = (unsigned*)ws;
  float* inv_norm = (float*)(ws + nbytes);
  float* agg      = (float*)(ws + 2 * nbytes);

  const int T = 256;

  k_zero_u32<<<(Nn + T - 1) / T, T, 0, stream>>>(deg, Nn);
  k_degree<<<(E + T - 1) / T, T, 0, stream>>>(src, deg, E);
  k_inv_norm<<<(Nn + T - 1) / T, T, 0, stream>>>(deg, inv_norm, Nn);

  const int total4 = Nn * (D_FEAT / 4);
  k_init_agg<<<(total4 + T - 1) / T, T, 0, stream>>>((const float4*)x, inv_norm,
                                                     (float4*)agg, total4);

  const long long sthreads = (long long)E * 32;
  k_scatter<<<(int)((sthreads + T - 1) / T), T, 0, stream>>>(
      (const float4*)x, src, dst, inv_norm, agg, E);

  k_gemm_wmma<<<(Nn + 15) / 16, T, 0, stream>>>(agg, inv_norm, W, bias, out, Nn);
}